// MyHingeLoss_cos_23888608101352
// MI455X (gfx1250) — compile-verified
//
#include <hip/hip_runtime.h>

// ---------------------------------------------------------------------------
// Cosine-embedding hinge loss, B=16384 rows x D=1024 f32.
// Memory-bound (~128MB HBM traffic; gather of target rows hits 192MB L2).
// One wave32 per row; coalesced b128 loads; cross-lane reduction done with
// V_WMMA_F32_16X16X4_F32 (A = lane values, B = ones -> row sums in D).
// ---------------------------------------------------------------------------

typedef float v2f __attribute__((ext_vector_type(2)));
typedef float v8f __attribute__((ext_vector_type(8)));

#define B_ROWS 16384
#define DIM    1024
#define WAVES_PER_BLOCK 8

// Reduce one f32 value across all 32 lanes of a wave using the f32 WMMA.
// A layout (ISA 7.12.2): lane l<16, VGPR v -> A[l][v]; lane 16+l, VGPR v -> A[l][2+v].
// With a = (val, 0): row m of A = [v_m, 0, v_{m+16}, 0].
// B = all ones (layout-independent): D[m][n] = v_m + v_{m+16}.
// C/D layout: lane l<16, VGPR r -> D[r][l%16]; lane >=16 -> D[8+r][l%16].
// Sum of a lane's 8 D regs = half-wave total; xor-16 shuffle add = full total.
__device__ __forceinline__ float wave_reduce_wmma(float v) {
    v2f a; a.x = v;    a.y = 0.0f;
    v2f b; b.x = 1.0f; b.y = 1.0f;
    v8f c = {};
    v8f d = __builtin_amdgcn_wmma_f32_16x16x4_f32(
        /*neg_a=*/false, a, /*neg_b=*/false, b,
        /*c_mod=*/(short)0, c, /*reuse_a=*/false, /*reuse_b=*/false);
    float s = d[0] + d[1] + d[2] + d[3] + d[4] + d[5] + d[6] + d[7];
    s += __shfl_xor(s, 16, 32);
    return s;
}

__global__ void __launch_bounds__(256)
zero_out_kernel(float* out) {
    if (threadIdx.x == 0 && blockIdx.x == 0) out[0] = 0.0f;
}

__global__ void __launch_bounds__(256)
hinge_cos_kernel(const float* __restrict__ output,
                 const float* __restrict__ target,
                 const int*   __restrict__ neg_idx,
                 float* __restrict__ out) {
    const int wave = threadIdx.x >> 5;       // 0..7
    const int lane = threadIdx.x & 31;
    const int row  = blockIdx.x * WAVES_PER_BLOCK + wave;   // grid exactly covers B_ROWS

    // negative row index: j = (i + 1 + neg) % B, guaranteed != i
    const int j = (row + 1 + neg_idx[row]) % B_ROWS;

    const float4* __restrict__ o4 = (const float4*)(output + (size_t)row * DIM);
    const float4* __restrict__ t4 = (const float4*)(target + (size_t)row * DIM);
    const float4* __restrict__ n4 = (const float4*)(target + (size_t)j   * DIM);

    float s_to = 0.0f;   // dot(t, o)
    float s_tt = 0.0f;   // ||t||^2
    float s_oo = 0.0f;   // ||o||^2
    float s_no = 0.0f;   // dot(t_neg, o)
    float s_nn = 0.0f;   // ||t_neg||^2

    // D/4 = 256 float4 per row; 32 lanes * 8 iterations, coalesced b128 loads.
    #pragma unroll
    for (int it = 0; it < DIM / 4 / 32; ++it) {
        const int idx = lane + 32 * it;
        const float4 ov = o4[idx];
        const float4 tv = t4[idx];
        const float4 nv = n4[idx];

        s_to = fmaf(tv.x, ov.x, s_to); s_to = fmaf(tv.y, ov.y, s_to);
        s_to = fmaf(tv.z, ov.z, s_to); s_to = fmaf(tv.w, ov.w, s_to);

        s_tt = fmaf(tv.x, tv.x, s_tt); s_tt = fmaf(tv.y, tv.y, s_tt);
        s_tt = fmaf(tv.z, tv.z, s_tt); s_tt = fmaf(tv.w, tv.w, s_tt);

        s_oo = fmaf(ov.x, ov.x, s_oo); s_oo = fmaf(ov.y, ov.y, s_oo);
        s_oo = fmaf(ov.z, ov.z, s_oo); s_oo = fmaf(ov.w, ov.w, s_oo);

        s_no = fmaf(nv.x, ov.x, s_no); s_no = fmaf(nv.y, ov.y, s_no);
        s_no = fmaf(nv.z, ov.z, s_no); s_no = fmaf(nv.w, ov.w, s_no);

        s_nn = fmaf(nv.x, nv.x, s_nn); s_nn = fmaf(nv.y, nv.y, s_nn);
        s_nn = fmaf(nv.z, nv.z, s_nn); s_nn = fmaf(nv.w, nv.w, s_nn);
    }

    // WMMA cross-lane reductions (EXEC is all-ones here: no divergence above).
    s_to = wave_reduce_wmma(s_to);
    s_tt = wave_reduce_wmma(s_tt);
    s_oo = wave_reduce_wmma(s_oo);
    s_no = wave_reduce_wmma(s_no);
    s_nn = wave_reduce_wmma(s_nn);

    __shared__ float warp_h[WAVES_PER_BLOCK];
    if (lane == 0) {
        const float eps = 1e-6f;
        const float den_p = fmaxf(sqrtf(s_tt) * sqrtf(s_oo), eps);
        const float den_n = fmaxf(sqrtf(s_nn) * sqrtf(s_oo), eps);
        const float cos_p = s_to / den_p;
        const float cos_n = s_no / den_n;
        warp_h[wave] = fmaxf(0.0f, 1.0f - cos_p + cos_n);
    }
    __syncthreads();

    if (threadIdx.x == 0) {
        float acc = 0.0f;
        #pragma unroll
        for (int w = 0; w < WAVES_PER_BLOCK; ++w) acc += warp_h[w];
        atomicAdd(out, acc * (1.0f / (float)B_ROWS));   // one atomic per block (2048 total)
    }
}

extern "C" void kernel_launch(void* const* d_in, const int* in_sizes, int n_in,
                              void* d_out, int out_size, void* d_ws, size_t ws_size,
                              hipStream_t stream) {
    const float* output  = (const float*)d_in[0];
    const float* target  = (const float*)d_in[1];
    const int*   neg_idx = (const int*)d_in[2];
    float* out = (float*)d_out;

    zero_out_kernel<<<1, 256, 0, stream>>>(out);

    const int blocks = B_ROWS / WAVES_PER_BLOCK;   // 2048
    hinge_cos_kernel<<<blocks, 256, 0, stream>>>(output, target, neg_idx, out);
}